// HGTSpatialEncoder_38500086841759
// MI455X (gfx1250) — compile-verified
//
#include <hip/hip_runtime.h>
#include <math.h>

// ---------------------------------------------------------------------------
// Types for CDNA5 WMMA f32 16x16x4
// ---------------------------------------------------------------------------
typedef __attribute__((ext_vector_type(2))) float v2f;
typedef __attribute__((ext_vector_type(8))) float v8f;

__device__ __forceinline__ float gelu_f(float x) {
    return 0.5f * x * (1.0f + erff(x * 0.70710678118654752f));
}

__device__ __forceinline__ void atomicMaxFloat(float* addr, float val) {
    if (val >= 0.0f)
        atomicMax((int*)addr, __float_as_int(val));
    else
        atomicMin((unsigned int*)addr, __float_as_uint(val));
}

__device__ __forceinline__ void atomicAddFloatFast(float* addr, float val) {
    __hip_atomic_fetch_add(addr, val, __ATOMIC_RELAXED, __HIP_MEMORY_SCOPE_AGENT);
}

// ---------------------------------------------------------------------------
// WMMA GEMM:  out[nrows x 128] = act(A[nrows x K]) @ W[K x 128] + bias
// EPI 0: plain store
// EPI 1: out = relu( g*val + (1-g)*xs_in ),  g = sigmoid(skip_ptr[skip_idx])
// ACT 0: identity     ACT 1: exact GELU applied to A on load
// Block = 256 threads = 8 waves; each wave: 16 rows x 128 cols (8 v8f accs).
//
// W is staged into LDS in fragment-interleaved order:
//   lds_w[(r>>1)*256 + c*2 + (r&1)] = W[r][c]
// so a B-fragment for (kstep k, lane-half lg, col) is ONE aligned ds_load_b64:
//   v2f b = *(v2f*)&lds_w[(2k+lg)*256 + col*2]  ->  (W[4k+2lg][col], W[4k+2lg+1][col])
// ---------------------------------------------------------------------------
template <int ACT, int EPI>
__global__ __launch_bounds__(256) void gemm_wmma(
    const float* __restrict__ A, const float* __restrict__ W,
    const float* __restrict__ bias, float* out,
    int nrows, int K,
    const float* __restrict__ skip_ptr, int skip_idx,
    const float* xs_in)
{
    __shared__ float lds_w[128 * 128];   // max K = 128 -> 64 KB

    const int tid = threadIdx.x;
    const int total = K * 128;
    for (int i = tid * 4; i < total; i += 256 * 4) {
        float4 w4 = *(const float4*)(W + i);
        const int r = i >> 7;            // source row (128 cols per row)
        const int c = i & 127;           // first of 4 consecutive cols
        const int base = (r >> 1) * 256 + (r & 1) + c * 2;
        lds_w[base + 0] = w4.x;
        lds_w[base + 2] = w4.y;
        lds_w[base + 4] = w4.z;
        lds_w[base + 6] = w4.w;
    }
    __syncthreads();

    const int wave = tid >> 5;
    const int lane = tid & 31;
    const int rowbase = blockIdx.x * 128 + wave * 16;
    if (rowbase >= nrows) return;   // nrows % 16 == 0: tiles are always full

    const int lh = lane & 15;       // row-within-tile / column selector
    const int lg = lane >> 4;       // lane half: K pair (0,1) vs (2,3)

    v8f acc[8] = {};

    const float* arow = A + (size_t)(rowbase + lh) * K + 2 * lg;
    const int ksteps = K >> 2;
    // LDS base for this lane: pair q = 2k+lg, col offset lh*2 added per tile
    const float* lbase = lds_w + lg * 256 + lh * 2;

    for (int k = 0; k < ksteps; ++k) {
        float2 av = *(const float2*)(arow + 4 * k);
        float ax = av.x, ay = av.y;
        if (ACT == 1) { ax = gelu_f(ax); ay = gelu_f(ay); }
        v2f a; a.x = ax; a.y = ay;

        const float* lk = lbase + k * 512;      // (2k+lg)*256 + lh*2
#pragma unroll
        for (int c = 0; c < 8; ++c) {
            v2f b = *(const v2f*)(lk + c * 32);  // one ds_load_b64
            acc[c] = __builtin_amdgcn_wmma_f32_16x16x4_f32(
                false, a, false, b, (short)0, acc[c], false, false);
        }
    }

    float gate = 1.0f;
    if (EPI == 1) {
        const float sv = skip_ptr[skip_idx];
        gate = 1.0f / (1.0f + expf(-sv));
    }

#pragma unroll
    for (int c = 0; c < 8; ++c) {
        const int col = c * 16 + lh;
        const float bv = bias[col];
#pragma unroll
        for (int g = 0; g < 8; ++g) {
            const int row = rowbase + (lg ? g + 8 : g);
            float val = acc[c][g] + bv;
            if (EPI == 1) {
                const float x0 = xs_in[(size_t)row * 128 + col];
                val = gate * val + (1.0f - gate) * x0;
                val = fmaxf(val, 0.0f);
            }
            out[(size_t)row * 128 + col] = val;
        }
    }
}

// ---------------------------------------------------------------------------
// Precompute relation-fused weights:
//   WkR[l,r] = Wk[l,src(r)] @ blockdiag(Krel[l,r,h])   (and bkR, WvR, bvR)
// ---------------------------------------------------------------------------
__global__ __launch_bounds__(256) void rel_combine(
    const float* __restrict__ Wk, const float* __restrict__ bk,
    const float* __restrict__ Wv, const float* __restrict__ bv,
    const float* __restrict__ Krel, const float* __restrict__ Vrel,
    float* WkR, float* bkR, float* WvR, float* bvR)
{
    const int lr = blockIdx.y;            // l*3 + r
    const int l = lr / 3, r = lr % 3;
    const int srcs[3] = {0, 1, 1};
    const int s = srcs[r];

    const int matsz = 129 * 128;          // 128 weight rows + 1 bias row
    int gid = blockIdx.x * 256 + threadIdx.x;
    if (gid >= 2 * matsz) return;
    const int mat = gid / matsz;          // 0 = K path, 1 = V path
    const int idx = gid % matsz;
    const int i = idx / 128;              // 0..127 weight row, 128 = bias
    const int j = idx % 128;
    const int h = j >> 5, e2 = j & 31;

    const float* Wsrc = (mat == 0 ? Wk : Wv) + (size_t)(l * 2 + s) * 128 * 128;
    const float* bsrc = (mat == 0 ? bk : bv) + (size_t)(l * 2 + s) * 128;
    const float* R = (mat == 0 ? Krel : Vrel) + (size_t)((l * 3 + r) * 4 + h) * 1024;

    float acc = 0.0f;
    if (i < 128) {
        for (int d = 0; d < 32; ++d)
            acc += Wsrc[i * 128 + h * 32 + d] * R[d * 32 + e2];
        (mat == 0 ? WkR : WvR)[(size_t)(l * 3 + r) * 16384 + i * 128 + j] = acc;
    } else {
        for (int d = 0; d < 32; ++d)
            acc += bsrc[h * 32 + d] * R[d * 32 + e2];
        (mat == 0 ? bkR : bvR)[(size_t)(l * 3 + r) * 128 + j] = acc;
    }
}

// ---------------------------------------------------------------------------
// Per-layer init: msg = 0, m = -inf, den = 0
// ---------------------------------------------------------------------------
__global__ __launch_bounds__(256) void init_layer(float* msg, float* m, float* den, int n)
{
    const int gid = blockIdx.x * 256 + threadIdx.x;
    if (gid < 2 * n * 128) msg[gid] = 0.0f;
    if (gid < 2 * n * 4) { m[gid] = -INFINITY; den[gid] = 0.0f; }
}

// ---------------------------------------------------------------------------
// Pass A: per (edge, head) score + running segment max (atomic)
// ---------------------------------------------------------------------------
__global__ __launch_bounds__(256) void edge_scores(
    const int* __restrict__ src, const int* __restrict__ dst, int E,
    const float* __restrict__ kt, const float* __restrict__ q,
    const float* __restrict__ prel4, float inv_sqrt_d,
    float* sc, float* m)
{
    const int gid = blockIdx.x * 256 + threadIdx.x;
    if (gid >= E * 4) return;
    const int e = gid >> 2, h = gid & 3;
    const int sn = src[e], dn = dst[e];

    const float4* kp = (const float4*)(kt + (size_t)sn * 128 + h * 32);
    const float4* qp = (const float4*)(q + (size_t)dn * 128 + h * 32);
    float dotv = 0.0f;
#pragma unroll
    for (int i = 0; i < 8; ++i) {
        float4 a = kp[i], b = qp[i];
        dotv += a.x * b.x + a.y * b.y + a.z * b.z + a.w * b.w;
    }
    const float s = dotv * prel4[h] * inv_sqrt_d;
    sc[gid] = s;
    atomicMaxFloat(m + (size_t)dn * 4 + h, s);
}

// ---------------------------------------------------------------------------
// Pass B: e = exp(sc - max); accumulate denominator
// ---------------------------------------------------------------------------
__global__ __launch_bounds__(256) void edge_expsum(
    const int* __restrict__ dst, int E,
    const float* __restrict__ m, float* sc, float* den)
{
    const int gid = blockIdx.x * 256 + threadIdx.x;
    if (gid >= E * 4) return;
    const int e = gid >> 2, h = gid & 3;
    const int dn = dst[e];
    const float v = expf(sc[gid] - m[(size_t)dn * 4 + h]);
    sc[gid] = v;
    atomicAddFloatFast(den + (size_t)dn * 4 + h, v);
}

// ---------------------------------------------------------------------------
// Pass C: msg[dst] += alpha * vt[src]   (128 threads per edge, coalesced)
// ---------------------------------------------------------------------------
__global__ __launch_bounds__(256) void edge_aggregate(
    const int* __restrict__ src, const int* __restrict__ dst, int E,
    const float* __restrict__ vt, const float* __restrict__ sc,
    const float* __restrict__ den, float* msg)
{
    const long long gid = (long long)blockIdx.x * 256 + threadIdx.x;
    if (gid >= (long long)E * 128) return;
    const int e = (int)(gid >> 7);
    const int c = (int)(gid & 127);
    const int h = c >> 5;
    const int sn = src[e], dn = dst[e];
    const float alpha = sc[(size_t)e * 4 + h] / den[(size_t)dn * 4 + h];
    atomicAddFloatFast(msg + (size_t)dn * 128 + c, alpha * vt[(size_t)sn * 128 + c]);
}

// ---------------------------------------------------------------------------
// Host-side orchestration
// ---------------------------------------------------------------------------
extern "C" void kernel_launch(void* const* d_in, const int* in_sizes, int n_in,
                              void* d_out, int out_size, void* d_ws, size_t ws_size,
                              hipStream_t stream)
{
    const float* x_region = (const float*)d_in[0];
    const float* x_site   = (const float*)d_in[1];
    const float* pWr      = (const float*)d_in[2];
    const float* pbr      = (const float*)d_in[3];
    const float* pWs      = (const float*)d_in[4];
    const float* pbs      = (const float*)d_in[5];
    const float* Wk       = (const float*)d_in[6];
    const float* bk       = (const float*)d_in[7];
    const float* Wq       = (const float*)d_in[8];
    const float* bq       = (const float*)d_in[9];
    const float* Wv       = (const float*)d_in[10];
    const float* bv       = (const float*)d_in[11];
    const float* Wo       = (const float*)d_in[12];
    const float* bo       = (const float*)d_in[13];
    const float* skip     = (const float*)d_in[14];
    const float* Krel     = (const float*)d_in[15];
    const float* Vrel     = (const float*)d_in[16];
    const float* prel     = (const float*)d_in[17];
    const int* edges[3] = { (const int*)d_in[18], (const int*)d_in[19], (const int*)d_in[20] };

    const int N = in_sizes[0] / 64;
    const int E = in_sizes[18] / 2;
    const float INV_SQRT_D = 0.17677669529663687f;   // 1/sqrt(32)

    float* out = (float*)d_out;
    float* xs[2] = { out, out + (size_t)N * 128 };

    // workspace carve-up
    float* w = (float*)d_ws;
    float* q    = w;  w += (size_t)2 * N * 128;
    float* ktb  = w;  w += (size_t)N * 128;
    float* vtb  = w;  w += (size_t)N * 128;
    float* msg  = w;  w += (size_t)2 * N * 128;
    float* scb  = w;  w += (size_t)3 * E * 4;
    float* mb   = w;  w += (size_t)2 * N * 4;
    float* denb = w;  w += (size_t)2 * N * 4;
    float* WkR  = w;  w += 6 * 16384;
    float* WvR  = w;  w += 6 * 16384;
    float* bkR  = w;  w += 6 * 128;
    float* bvR  = w;  w += 6 * 128;
    (void)n_in; (void)out_size; (void)ws_size;

    const dim3 blk(256);
    const int gproj = (N + 127) / 128;
    const int gedge4 = (E * 4 + 255) / 256;
    const long long nagg = (long long)E * 128;
    const int gagg = (int)((nagg + 255) / 256);
    const int ginit = (2 * N * 128 + 255) / 256;

    const int srcs[3] = {0, 1, 1}, dsts[3] = {1, 0, 1};

    // Relation-fused weight precompute (both layers, K and V paths)
    rel_combine<<<dim3((2 * 129 * 128 + 255) / 256, 6), blk, 0, stream>>>(
        Wk, bk, Wv, bv, Krel, Vrel, WkR, bkR, WvR, bvR);

    // Input projections -> xs (stored in d_out, updated in place per layer)
    gemm_wmma<0, 0><<<gproj, blk, 0, stream>>>(x_region, pWr, pbr, xs[0], N, 64, nullptr, 0, nullptr);
    gemm_wmma<0, 0><<<gproj, blk, 0, stream>>>(x_site,   pWs, pbs, xs[1], N, 32, nullptr, 0, nullptr);

    for (int l = 0; l < 2; ++l) {
        init_layer<<<ginit, blk, 0, stream>>>(msg, mb, denb, N);

        // q projections per destination node type
        for (int t = 0; t < 2; ++t)
            gemm_wmma<0, 0><<<gproj, blk, 0, stream>>>(
                xs[t], Wq + (size_t)(l * 2 + t) * 16384, bq + (size_t)(l * 2 + t) * 128,
                q + (size_t)t * N * 128, N, 128, nullptr, 0, nullptr);

        // Pass A: kt GEMM + scores + segment max (kt buffer reused across r)
        for (int r = 0; r < 3; ++r) {
            const int s = srcs[r], d = dsts[r];
            gemm_wmma<0, 0><<<gproj, blk, 0, stream>>>(
                xs[s], WkR + (size_t)(l * 3 + r) * 16384, bkR + (size_t)(l * 3 + r) * 128,
                ktb, N, 128, nullptr, 0, nullptr);
            edge_scores<<<gedge4, blk, 0, stream>>>(
                edges[r], edges[r] + E, E, ktb, q + (size_t)d * N * 128,
                prel + (size_t)(l * 3 + r) * 4, INV_SQRT_D,
                scb + (size_t)r * E * 4, mb + (size_t)d * N * 4);
        }

        // Pass B: exp + denominator (after all maxes in each bucket are done)
        for (int r = 0; r < 3; ++r) {
            const int d = dsts[r];
            edge_expsum<<<gedge4, blk, 0, stream>>>(
                edges[r] + E, E, mb + (size_t)d * N * 4,
                scb + (size_t)r * E * 4, denb + (size_t)d * N * 4);
        }

        // Pass C: vt GEMM + weighted scatter into msg (vt buffer reused)
        for (int r = 0; r < 3; ++r) {
            const int s = srcs[r], d = dsts[r];
            gemm_wmma<0, 0><<<gproj, blk, 0, stream>>>(
                xs[s], WvR + (size_t)(l * 3 + r) * 16384, bvR + (size_t)(l * 3 + r) * 128,
                vtb, N, 128, nullptr, 0, nullptr);
            edge_aggregate<<<gagg, blk, 0, stream>>>(
                edges[r], edges[r] + E, E, vtb,
                scb + (size_t)r * E * 4, denb + (size_t)d * N * 4,
                msg + (size_t)d * N * 128);
        }

        // Output: gelu(msg) @ Wo + bo, skip gate, relu — in place on xs[t]
        for (int t = 0; t < 2; ++t)
            gemm_wmma<1, 1><<<gproj, blk, 0, stream>>>(
                msg + (size_t)t * N * 128,
                Wo + (size_t)(l * 2 + t) * 16384, bo + (size_t)(l * 2 + t) * 128,
                xs[t], N, 128, skip, l * 2 + t, xs[t]);
    }
}